// DynamicGNN_81295140978882
// MI455X (gfx1250) — compile-verified
//
#include <hip/hip_runtime.h>
#include <math.h>

// ---- problem constants (match reference) ----
constexpr int kB    = 64;
constexpr int kN    = 1024;
constexpr int kNODE = 32;
constexpr int kHID  = 64;
constexpr float kTAU = 2.0f;

typedef __attribute__((ext_vector_type(2))) float v2f;
typedef __attribute__((ext_vector_type(8))) float v8f;

// LDS layout (floats)
constexpr int HSTRIDE = kN + 4;   // 1028: bank-conflict-free b64 column reads (4n+j covers 64 banks)
constexpr int MSTRIDE = 36;       // m-tile row stride: 16B aligned, 36n mod 64 distinct for n=0..15
constexpr int SMEM_FLOATS =
    4 * kN                        // sQ: float4 per node (qx, qy, qc, pad)
  + kNODE * HSTRIDE               // sHT: transposed h slab [32][1028]
  + kHID * kNODE                  // sW1T: W1 transposed [64][32]
  + kHID * kNODE                  // sW2 [64][32]
  + kHID + kNODE                  // sB1, sB2
  + 8 * 16 * MSTRIDE;             // sM: per-wave 16x32 m tiles

// ============================================================
// Kernel 1: h0[b,n,c] = past[b,n,:] @ embed_w + embed_b
// ============================================================
__global__ __launch_bounds__(256) void k_embed(
    const float* __restrict__ past, const float* __restrict__ ew,
    const float* __restrict__ eb, float* __restrict__ h0)
{
  int idx = blockIdx.x * 256 + threadIdx.x;   // B*N*NODE threads
  int c  = idx & 31;
  int bn = idx >> 5;
  float px = past[bn * 2 + 0];
  float py = past[bn * 2 + 1];
  h0[idx] = fmaf(px, ew[c], fmaf(py, ew[32 + c], eb[c]));
}

// ============================================================
// Kernel 2: one GNN layer, fused softmax-attention (WMMA) + MLP
//   hout = hin + relu((softmax(-d2/tau) @ hin) @ W1 + b1) @ W2 + b2
// Grid: kB * 8 blocks (batch x row-group of 128), 256 threads (8 waves),
// each wave owns 16 rows.
// ============================================================
__global__ __launch_bounds__(256) void k_gnn(
    const float* __restrict__ past,
    const float* __restrict__ hin, float* __restrict__ hout,
    const float* __restrict__ w1, const float* __restrict__ b1,
    const float* __restrict__ w2, const float* __restrict__ b2)
{
  extern __shared__ float smem[];
  float4* sQ   = (float4*)smem;                     // [kN]
  float*  sHT  = smem + 4 * kN;                     // [kNODE][HSTRIDE]
  float*  sW1T = sHT + kNODE * HSTRIDE;             // [64][32]  (W1 transposed)
  float*  sW2  = sW1T + kHID * kNODE;               // [64][32]
  float*  sB1  = sW2 + kHID * kNODE;                // [64]
  float*  sB2  = sB1 + kHID;                        // [32]
  float*  sM   = sB2 + kNODE;                       // [8][16][MSTRIDE]

  const int t  = threadIdx.x;
  const int bb = blockIdx.x >> 3;
  const int rg = blockIdx.x & 7;

  const float LOG2E = 1.4426950408889634f;
  const float kq = (2.0f / kTAU) * LOG2E;

  // ---- cooperative load phase ----
  for (int j = t; j < kN; j += 256) {
    float x = past[(bb * kN + j) * 2 + 0];
    float y = past[(bb * kN + j) * 2 + 1];
    // attention logit (row-constant factored out; cancels in softmax):
    //   log2 w_ij = xi*qx + yi*qy + qc
    sQ[j] = make_float4(x * kq, y * kq, -(x * x + y * y) * (LOG2E / kTAU), 0.0f);
  }
  const float* hsrc = hin + (size_t)bb * kN * kNODE;
  for (int i = t; i < kN * kNODE; i += 256) {       // coalesced read, transposed LDS store
    int j = i >> 5, c = i & 31;
    sHT[c * HSTRIDE + j] = hsrc[i];
  }
  for (int i = t; i < kNODE * kHID; i += 256) {     // W1[k][h] -> sW1T[h][k]
    int k = i >> 6, h = i & 63;
    sW1T[h * kNODE + k] = w1[i];
  }
  for (int i = t; i < kHID * kNODE; i += 256) sW2[i] = w2[i];
  if (t < kHID)              sB1[t] = b1[t];
  else if (t < kHID + kNODE) sB2[t - kHID] = b2[t - kHID];
  __syncthreads();

  const int wave = t >> 5;
  const int lane = t & 31;
  const int hv   = lane >> 4;            // half-wave: selects K pair (A) / K rows (B)
  const int n    = lane & 15;
  const int rowBase = rg * 128 + wave * 16;
  const int rowG = rowBase + n;          // this lane's attention row (A-matrix M index)

  const float xi = past[(bb * kN + rowG) * 2 + 0];
  const float yi = past[(bb * kN + rowG) * 2 + 1];

  v8f c0 = {};                            // m-tile cols 0..15, f32 accum
  v8f c1 = {};                            // m-tile cols 16..31
  float sacc = 0.0f;                      // partial softmax denominator

  const float* hcol_lo = sHT + n * HSTRIDE;
  const float* hcol_hi = sHT + (n + 16) * HSTRIDE;

  // ---- main contraction: m = softmax-weights @ h, K chunks of 4 ----
  for (int jc = 0; jc < kN; jc += 4) {
    const int js = jc + hv * 2;           // lanes 0-15: K=0,1  lanes 16-31: K=2,3
    float4 q0 = sQ[js];
    float4 q1 = sQ[js + 1];
    // raw v_exp_f32: exponents are small, no range fixup needed
    float e0 = __builtin_amdgcn_exp2f(fmaf(xi, q0.x, fmaf(yi, q0.y, q0.z)));
    float e1 = __builtin_amdgcn_exp2f(fmaf(xi, q1.x, fmaf(yi, q1.y, q1.z)));
    sacc += e0 + e1;
    v2f a; a[0] = e0; a[1] = e1;          // A frag: 16x4 f32 layout
    const v2f bl = *(const v2f*)(hcol_lo + js);   // B frag: h[js..js+1][n]
    const v2f bh = *(const v2f*)(hcol_hi + js);   // B frag: h[js..js+1][n+16]
    c0 = __builtin_amdgcn_wmma_f32_16x16x4_f32(false, a, false, bl, (short)0, c0, false, false);
    c1 = __builtin_amdgcn_wmma_f32_16x16x4_f32(false, a, false, bh, (short)0, c1, false, false);
  }

  // ---- softmax normalization: combine half-wave partial sums ----
  float ssum = sacc + __shfl_xor(sacc, 16, 32);  // lane L now has full sum of row (L&15)

  // scale C-frags by 1/rowsum and spill m-tile to LDS
  float* sMw = sM + wave * 16 * MSTRIDE;
#pragma unroll
  for (int r = 0; r < 8; ++r) {
    const int m = hv * 8 + r;                    // C layout: VGPR r -> M = r + 8*half
    const float inv = 1.0f / __shfl(ssum, m, 32);
    sMw[m * MSTRIDE + n]      = c0[r] * inv;
    sMw[m * MSTRIDE + n + 16] = c1[r] * inv;
  }

  // ---- MLP: row = lane&15, hidden half = hv*32 .. hv*32+31 ----
  float mr[kNODE];
  {
    const float4* mp = (const float4*)(sMw + n * MSTRIDE);
#pragma unroll
    for (int k4 = 0; k4 < 8; ++k4) {
      float4 v = mp[k4];
      mr[4*k4+0] = v.x; mr[4*k4+1] = v.y; mr[4*k4+2] = v.z; mr[4*k4+3] = v.w;
    }
  }
  const int hb = hv * 32;
  float po[kNODE];
#pragma unroll
  for (int c = 0; c < kNODE; ++c) po[c] = 0.0f;
  for (int hh = 0; hh < 32; ++hh) {              // rolled; inner loops unrolled -> regs
    float acc = sB1[hb + hh];
    const float4* w1row = (const float4*)(sW1T + (hb + hh) * kNODE);
#pragma unroll
    for (int k4 = 0; k4 < 8; ++k4) {             // 8x ds_load_b128 (broadcast)
      float4 wv = w1row[k4];
      acc = fmaf(mr[4*k4+0], wv.x, acc);
      acc = fmaf(mr[4*k4+1], wv.y, acc);
      acc = fmaf(mr[4*k4+2], wv.z, acc);
      acc = fmaf(mr[4*k4+3], wv.w, acc);
    }
    acc = fmaxf(acc, 0.0f);                      // relu
    const float4* w2row = (const float4*)(sW2 + (hb + hh) * kNODE);
#pragma unroll
    for (int c4 = 0; c4 < 8; ++c4) {             // 8x ds_load_b128 (broadcast)
      float4 wv = w2row[c4];
      po[4*c4+0] = fmaf(acc, wv.x, po[4*c4+0]);
      po[4*c4+1] = fmaf(acc, wv.y, po[4*c4+1]);
      po[4*c4+2] = fmaf(acc, wv.z, po[4*c4+2]);
      po[4*c4+3] = fmaf(acc, wv.w, po[4*c4+3]);
    }
  }

  // combine hidden halves, add bias + residual, store as float4
  float tot[kNODE];
#pragma unroll
  for (int c = 0; c < kNODE; ++c) {              // shuffles executed by all lanes
    tot[c] = po[c] + __shfl_xor(po[c], 16, 32) + sB2[c] + sHT[c * HSTRIDE + rowG];
  }
  if (hv == 0) {
    float4* orow = (float4*)(hout + ((size_t)bb * kN + rowG) * kNODE);
#pragma unroll
    for (int c4 = 0; c4 < 8; ++c4)
      orow[c4] = make_float4(tot[4*c4+0], tot[4*c4+1], tot[4*c4+2], tot[4*c4+3]);
  }
}

// ============================================================
// Kernel 3: mean over nodes + readout MLP. One block per batch.
// ============================================================
__global__ __launch_bounds__(256) void k_readout(
    const float* __restrict__ h,
    const float* __restrict__ w1, const float* __restrict__ b1,
    const float* __restrict__ w2, const float* __restrict__ b2,
    float* __restrict__ out)
{
  __shared__ float sRed[8][kNODE];
  __shared__ float sG[kNODE];
  __shared__ float sA[kHID];
  const int bb = blockIdx.x, t = threadIdx.x;
  const int c = t & 31, js = t >> 5;
  float acc = 0.0f;
  for (int j = js; j < kN; j += 8) acc += h[((size_t)bb * kN + j) * kNODE + c];
  sRed[js][c] = acc;
  __syncthreads();
  if (t < kNODE) {
    float g = 0.0f;
#pragma unroll
    for (int r = 0; r < 8; ++r) g += sRed[r][t];
    sG[t] = g * (1.0f / kN);
  }
  __syncthreads();
  if (t < kHID) {
    float a = b1[t];
#pragma unroll
    for (int k = 0; k < kNODE; ++k) a = fmaf(sG[k], w1[k * kHID + t], a);
    sA[t] = fmaxf(a, 0.0f);
  }
  __syncthreads();
  if (t < kHID) {
    float o = b2[t];
#pragma unroll
    for (int hh = 0; hh < kHID; ++hh) o = fmaf(sA[hh], w2[hh * kHID + t], o);
    out[bb * kHID + t] = o;
  }
}

// ============================================================
extern "C" void kernel_launch(void* const* d_in, const int* in_sizes, int n_in,
                              void* d_out, int out_size, void* d_ws, size_t ws_size,
                              hipStream_t stream) {
  const float* past    = (const float*)d_in[0];
  const float* embed_w = (const float*)d_in[1];
  const float* embed_b = (const float*)d_in[2];
  const float* mlp_w1  = (const float*)d_in[3];   // [2,32,64]
  const float* mlp_b1  = (const float*)d_in[4];   // [2,64]
  const float* mlp_w2  = (const float*)d_in[5];   // [2,64,32]
  const float* mlp_b2  = (const float*)d_in[6];   // [2,32]
  const float* ro_w1   = (const float*)d_in[7];
  const float* ro_b1   = (const float*)d_in[8];
  const float* ro_w2   = (const float*)d_in[9];
  const float* ro_b2   = (const float*)d_in[10];

  float* hA = (float*)d_ws;                        // [B,N,NODE] ping
  float* hB = hA + (size_t)kB * kN * kNODE;        // [B,N,NODE] pong

  k_embed<<<(kB * kN * kNODE) / 256, 256, 0, stream>>>(past, embed_w, embed_b, hA);

  const size_t shmem = (size_t)SMEM_FLOATS * sizeof(float);   // ~179 KB of the 320 KB LDS
  hipFuncSetAttribute((const void*)k_gnn,
                      hipFuncAttributeMaxDynamicSharedMemorySize, (int)shmem);

  // layer 0: hA -> hB ; layer 1: hB -> hA
  k_gnn<<<kB * 8, 256, shmem, stream>>>(past, hA, hB,
      mlp_w1, mlp_b1, mlp_w2, mlp_b2);
  k_gnn<<<kB * 8, 256, shmem, stream>>>(past, hB, hA,
      mlp_w1 + kNODE * kHID, mlp_b1 + kHID, mlp_w2 + kHID * kNODE, mlp_b2 + kNODE);

  k_readout<<<kB, 256, 0, stream>>>(hA, ro_w1, ro_b1, ro_w2, ro_b2, (float*)d_out);
}